// BaseAttention_2216203125461
// MI455X (gfx1250) — compile-verified
//
#include <hip/hip_runtime.h>
#include <hip/hip_bf16.h>
#include <math.h>

#define RNN_SIZE 2048
#define ATT_HID  512
#define BATCH    128
#define LSEQ     196

typedef float v2f __attribute__((ext_vector_type(2)));
typedef float v8f __attribute__((ext_vector_type(8)));

// ---------------------------------------------------------------------------
// Kernel 1: att_h[b][a] = sum_k h[b][k] * W[a][k] + bias[a]
//   h: [128, 2048] row-major, W: [512, 2048] row-major  -> att_h: [128, 512]
// One wave32 per 16x16 output tile, K marched in steps of 4 using
// V_WMMA_F32_16X16X4_F32 (exact f32 accumulate, no precision loss).
//
// ISA 32-bit A 16x4 layout:  lanes 0-15 hold K=0,1 in v0,v1 (M = lane);
//                            lanes 16-31 hold K=2,3 (M = lane-16).
// B 4x16 mirrors it with N striped across lanes.
// C/D 16x16 f32: lane%16 = N, VGPR i holds M = i + 8*(lane/16).
// ---------------------------------------------------------------------------
__global__ __launch_bounds__(256)
void atth_gemm_wmma(const float* __restrict__ h,
                    const float* __restrict__ w,      // [ATT_HID, RNN_SIZE]
                    const float* __restrict__ bias,   // [ATT_HID]
                    float* __restrict__ att_h)        // [BATCH, ATT_HID]
{
    const int lane = threadIdx.x & 31;
    const int wave = threadIdx.x >> 5;
    const int tile = blockIdx.x * 8 + wave;     // 32 blocks * 8 waves = 256 tiles
    const int tn = (tile & 31) << 4;            // N tile base: 0..511 step 16
    const int tm = (tile >> 5) << 4;            // M tile base: 0..127 step 16
    const int half = lane >> 4;                 // 0: K 0,1   1: K 2,3
    const int lr   = lane & 15;

    const float* aRow = h + (size_t)(tm + lr) * RNN_SIZE + (half << 1);
    const float* bCol = w + (size_t)(tn + lr) * RNN_SIZE + (half << 1);

    v8f c = {};
#pragma unroll 8
    for (int k = 0; k < RNN_SIZE; k += 4) {
        v2f a  = *(const v2f*)(aRow + k);       // A[M=lr][k+klo .. k+klo+1]
        v2f bb = *(const v2f*)(bCol + k);       // B[k+klo .. ][N=lr] == W[tn+lr][k+klo..]
        c = __builtin_amdgcn_wmma_f32_16x16x4_f32(
                /*neg_a=*/false, a, /*neg_b=*/false, bb,
                /*c_mod=*/(short)0, c, /*reuse_a=*/false, /*reuse_b=*/false);
    }

    const int col = tn + lr;
    const float bv = bias[col];
#pragma unroll
    for (int i = 0; i < 8; ++i) {
        const int row = tm + (half << 3) + i;
        att_h[(size_t)row * ATT_HID + col] = c[i] + bv;
    }
}

// ---------------------------------------------------------------------------
// Kernel 2: fused  scores -> softmax -> weighted sum, one block per batch b.
//   scores[l] = alpha_b + sum_a tanh(internal[b][l][a] + att_h[b][a]) * alpha_w[a]
//   weight    = softmax_l(scores)
//   out[b][d] = sum_l weight[l] * att_feats[b][l][d]
// Streams internal (51 MB) and att_feats (205 MB) exactly once, coalesced.
// ---------------------------------------------------------------------------
__global__ __launch_bounds__(256)
void attn_fused(const float* __restrict__ att_h,      // [BATCH, ATT_HID]
                const float* __restrict__ internal,   // [BATCH, LSEQ, ATT_HID]
                const float* __restrict__ att_feats,  // [BATCH, LSEQ, RNN_SIZE]
                const float* __restrict__ alpha_w,    // [ATT_HID]
                const float* __restrict__ alpha_b,    // [1]
                float* __restrict__ out)              // [BATCH, RNN_SIZE]
{
    __shared__ float s_atth[ATT_HID];
    __shared__ float s_aw[ATT_HID];
    __shared__ float s_wgt[LSEQ + 12];   // scores, then softmax weights

    const int b    = blockIdx.x;
    const int tid  = threadIdx.x;        // 256 threads = 8 waves
    const int lane = tid & 31;
    const int wave = tid >> 5;

    for (int i = tid; i < ATT_HID; i += 256) {
        s_atth[i] = att_h[(size_t)b * ATT_HID + i];
        s_aw[i]   = alpha_w[i];
    }
    __syncthreads();

    const float ab = alpha_b[0];

    // --- scores: one wave per l, lanes strided over the hidden dim ---
    for (int l = wave; l < LSEQ; l += 8) {
        const float* row = internal + ((size_t)b * LSEQ + l) * ATT_HID;
        float s = 0.f;
#pragma unroll 4
        for (int a = lane; a < ATT_HID; a += 32)
            s += tanhf(row[a] + s_atth[a]) * s_aw[a];
#pragma unroll
        for (int off = 16; off > 0; off >>= 1)
            s += __shfl_xor(s, off, 32);
        if (lane == 0) s_wgt[l] = s + ab;
    }
    __syncthreads();

    // --- softmax over l (wave 0) ---
    if (wave == 0) {
        float m = -INFINITY;
        for (int l = lane; l < LSEQ; l += 32) m = fmaxf(m, s_wgt[l]);
#pragma unroll
        for (int off = 16; off > 0; off >>= 1)
            m = fmaxf(m, __shfl_xor(m, off, 32));
        float sum = 0.f;
        for (int l = lane; l < LSEQ; l += 32) {
            float e = __expf(s_wgt[l] - m);
            s_wgt[l] = e;
            sum += e;
        }
#pragma unroll
        for (int off = 16; off > 0; off >>= 1)
            sum += __shfl_xor(sum, off, 32);
        const float inv = 1.0f / sum;
        for (int l = lane; l < LSEQ; l += 32) s_wgt[l] *= inv;
    }
    __syncthreads();

    // --- weighted sum over att_feats, coalesced across threads ---
    const float* base = att_feats + (size_t)b * LSEQ * RNN_SIZE;
#pragma unroll
    for (int d = tid; d < RNN_SIZE; d += 256) {
        float acc = 0.f;
        const float* p = base + d;
#pragma unroll 4
        for (int l = 0; l < LSEQ; ++l)
            acc += s_wgt[l] * p[(size_t)l * RNN_SIZE];
        out[(size_t)b * RNN_SIZE + d] = acc;
    }
}

// ---------------------------------------------------------------------------
extern "C" void kernel_launch(void* const* d_in, const int* in_sizes, int n_in,
                              void* d_out, int out_size, void* d_ws, size_t ws_size,
                              hipStream_t stream) {
    const float* h          = (const float*)d_in[0];   // [128, 2048]
    const float* att_feats  = (const float*)d_in[1];   // [128, 196, 2048]
    const float* internal   = (const float*)d_in[2];   // [128, 196, 512]
    const float* h2att_w    = (const float*)d_in[3];   // [512, 2048]
    const float* h2att_b    = (const float*)d_in[4];   // [512]
    const float* alpha_w    = (const float*)d_in[5];   // [512]
    const float* alpha_b    = (const float*)d_in[6];   // [1]
    float* out = (float*)d_out;                        // [128, 2048]

    float* att_h = (float*)d_ws;                       // 128*512*4 = 256 KB scratch

    // 256 tiles (8 M-tiles x 32 N-tiles), 8 waves per block
    atth_gemm_wmma<<<32, 256, 0, stream>>>(h, h2att_w, h2att_b, att_h);

    // one block per batch row
    attn_fused<<<BATCH, 256, 0, stream>>>(att_h, internal, att_feats,
                                          alpha_w, alpha_b, out);
}